// myGATConv_s_13975823582067
// MI455X (gfx1250) — compile-verified
//
#include <hip/hip_runtime.h>

#define N_NODES   50000
#define N_EDGES   800000
#define IN_FEATS  256
#define OUT_FEATS 32
#define NUM_HEADS 8
#define HD        (NUM_HEADS * OUT_FEATS)   // 256

typedef __attribute__((ext_vector_type(2))) float v2f;
typedef __attribute__((ext_vector_type(8))) float v8f;

// -------- init: zero rst region of d_out, zero denom/attn_sum, set max-keys to 0 --------
__global__ __launch_bounds__(256) void gat_init(unsigned* __restrict__ mmax,
                                                float* __restrict__ denom,
                                                float* __restrict__ asum,
                                                float* __restrict__ rst) {
    int i = blockIdx.x * 256 + threadIdx.x;
    if (i < N_NODES * HD) rst[i] = 0.0f;
    if (i < N_NODES * NUM_HEADS) {
        mmax[i]  = 0u;      // encodes -inf under monotone float->uint key
        denom[i] = 0.0f;
        asum[i]  = 0.0f;
    }
}

// -------- GEMM: fs[N,256] = feat[N,256] @ fc_w[256,256]^T  via V_WMMA_F32_16X16X4_F32 ----
// grid (3125, 4), block 128 (4 waves). Block computes rows [16*bx,16*bx+16) x cols [64*by,64*by+64).
// Wave w handles the 16x16 tile at cols 64*by + 16*w. K chunked by 64 through LDS.
__global__ __launch_bounds__(128) void gat_gemm_wmma(const float* __restrict__ feat,
                                                     const float* __restrict__ fc_w,
                                                     float* __restrict__ fs) {
    constexpr int KC  = 64;
    constexpr int LDA = KC + 2;   // stride 66: 66 % 64 == 2 -> bank-conflict-free, keeps 8B align
    __shared__ float lds_a[16 * LDA];
    __shared__ float lds_b[64 * LDA];

    const int tid  = threadIdx.x;
    const int m0   = blockIdx.x * 16;
    const int n0b  = blockIdx.y * 64;
    const int wave = tid >> 5;
    const int lane = tid & 31;
    const int rc    = lane & 15;          // A row / B col / D col
    const int khalf = (lane >> 4) * 2;    // this lane's K pair within the 4-wide K step
    const int mhalf = (lane >> 4) * 8;    // D-matrix row half

    v8f acc = {};

    for (int kb = 0; kb < IN_FEATS; kb += KC) {
        // stage A chunk: 16 x 64 floats, coalesced float4 global loads
        for (int idx = tid * 4; idx < 16 * KC; idx += 128 * 4) {
            const int r = idx >> 6, k = idx & (KC - 1);
            const float4 v = *(const float4*)&feat[(size_t)(m0 + r) * IN_FEATS + kb + k];
            lds_a[r * LDA + k + 0] = v.x; lds_a[r * LDA + k + 1] = v.y;
            lds_a[r * LDA + k + 2] = v.z; lds_a[r * LDA + k + 3] = v.w;
        }
        // stage B chunk: 64 output-cols x 64 k. B[k][n] = fc_w[n][k] (fc_w row-major => coalesced)
        for (int idx = tid * 4; idx < 64 * KC; idx += 128 * 4) {
            const int r = idx >> 6, k = idx & (KC - 1);
            const float4 v = *(const float4*)&fc_w[(size_t)(n0b + r) * IN_FEATS + kb + k];
            lds_b[r * LDA + k + 0] = v.x; lds_b[r * LDA + k + 1] = v.y;
            lds_b[r * LDA + k + 2] = v.z; lds_b[r * LDA + k + 3] = v.w;
        }
        __syncthreads();

        const float* ap = &lds_a[rc * LDA + khalf];
        const float* bp = &lds_b[(wave * 16 + rc) * LDA + khalf];
#pragma unroll
        for (int kk = 0; kk < KC / 4; ++kk) {
            v2f a = *(const v2f*)(ap + kk * 4);   // ds_load_b64: K = kk*4+khalf, +1
            v2f b = *(const v2f*)(bp + kk * 4);
            acc = __builtin_amdgcn_wmma_f32_16x16x4_f32(
                /*neg_a=*/false, a, /*neg_b=*/false, b,
                /*c_mod=*/(short)0, acc, /*reuse_a=*/false, /*reuse_b=*/false);
        }
        __syncthreads();
    }

    // D layout: VGPR i -> lanes 0-15: (M=i, N=lane); lanes 16-31: (M=i+8, N=lane-16)
    const int n0 = n0b + wave * 16;
#pragma unroll
    for (int i = 0; i < 8; ++i)
        fs[(size_t)(m0 + mhalf + i) * HD + n0 + rc] = acc[i];
}

// -------- per-(edge,head) segment max over dst via monotone-key u32 atomicMax --------
__global__ __launch_bounds__(256) void gat_edge_max(const float* __restrict__ edge_emb,
                                                    const int* __restrict__ e_feat,
                                                    const int* __restrict__ dst,
                                                    unsigned* __restrict__ mmax) {
    int i = blockIdx.x * 256 + threadIdx.x;
    if (i >= N_EDGES * NUM_HEADS) return;
    const int e = i >> 3, h = i & 7;
    const float v = edge_emb[e_feat[e] * NUM_HEADS + h];
    const unsigned u = __float_as_uint(v);
    const unsigned key = (u & 0x80000000u) ? ~u : (u | 0x80000000u);
    atomicMax(&mmax[dst[e] * NUM_HEADS + h], key);
}

// -------- ex = exp(e - m[dst]); denom += ex; stash ex in attn output slot --------
__global__ __launch_bounds__(256) void gat_edge_exp(const float* __restrict__ edge_emb,
                                                    const int* __restrict__ e_feat,
                                                    const int* __restrict__ dst,
                                                    const unsigned* __restrict__ mmax,
                                                    float* __restrict__ denom,
                                                    float* __restrict__ attn) {
    int i = blockIdx.x * 256 + threadIdx.x;
    if (i >= N_EDGES * NUM_HEADS) return;
    const int e = i >> 3, h = i & 7;
    const int d = dst[e];
    const unsigned k = mmax[d * NUM_HEADS + h];
    const float m = __uint_as_float((k & 0x80000000u) ? (k & 0x7fffffffu) : ~k);
    const float ex = __expf(edge_emb[e_feat[e] * NUM_HEADS + h] - m);
    attn[i] = ex;
    atomicAdd(&denom[d * NUM_HEADS + h], ex);
}

// -------- attn = ex / denom[dst]; attn_sum += attn --------
__global__ __launch_bounds__(256) void gat_edge_norm(const int* __restrict__ dst,
                                                     const float* __restrict__ denom,
                                                     float* __restrict__ attn,
                                                     float* __restrict__ asum) {
    int i = blockIdx.x * 256 + threadIdx.x;
    if (i >= N_EDGES * NUM_HEADS) return;
    const int e = i >> 3, h = i & 7;
    const int d = dst[e];
    const float a = attn[i] / denom[d * NUM_HEADS + h];
    attn[i] = a;
    atomicAdd(&asum[d * NUM_HEADS + h], a);
}

// -------- rst[dst] += attn[e,h] * fs[src]  (64 threads/edge, float4 reads) --------
__global__ __launch_bounds__(256) void gat_aggregate(const int* __restrict__ src,
                                                     const int* __restrict__ dst,
                                                     const float* __restrict__ attn,
                                                     const float* __restrict__ fs,
                                                     float* __restrict__ rst) {
    int i = blockIdx.x * 256 + threadIdx.x;            // E*64 = 51,200,000 < 2^31
    if (i >= N_EDGES * (HD / 4)) return;
    const int e = i >> 6;
    const int j = (i & 63) * 4;                        // feature offset in [0,256), step 4
    const int h = j >> 5;                              // head
    const float a = attn[e * NUM_HEADS + h];
    const float4 v = *(const float4*)&fs[(size_t)src[e] * HD + j];
    float* o = &rst[(size_t)dst[e] * HD + j];
    atomicAdd(o + 0, a * v.x);
    atomicAdd(o + 1, a * v.y);
    atomicAdd(o + 2, a * v.z);
    atomicAdd(o + 3, a * v.w);
}

// -------- rst /= (attn_sum + 1e-12) --------
__global__ __launch_bounds__(256) void gat_finalize(float* __restrict__ rst,
                                                    const float* __restrict__ asum) {
    int i = blockIdx.x * 256 + threadIdx.x;
    if (i >= N_NODES * HD) return;
    const int n = i >> 8;
    const int h = (i >> 5) & 7;
    rst[i] = rst[i] / (asum[n * NUM_HEADS + h] + 1e-12f);
}

extern "C" void kernel_launch(void* const* d_in, const int* in_sizes, int n_in,
                              void* d_out, int out_size, void* d_ws, size_t ws_size,
                              hipStream_t stream) {
    (void)in_sizes; (void)n_in; (void)out_size; (void)ws_size;
    const float* feat     = (const float*)d_in[0];
    const float* fc_w     = (const float*)d_in[1];
    const float* edge_emb = (const float*)d_in[2];
    const int*   e_feat   = (const int*)d_in[3];
    const int*   src      = (const int*)d_in[4];
    const int*   dst      = (const int*)d_in[5];

    float* rst  = (float*)d_out;                         // [N, H, D]
    float* attn = rst + (size_t)N_NODES * HD;            // [E, H]

    float*    fs    = (float*)d_ws;                      // [N, 256] = 51.2 MB
    unsigned* mmax  = (unsigned*)(fs + (size_t)N_NODES * HD);
    float*    denom = (float*)(mmax + (size_t)N_NODES * NUM_HEADS);
    float*    asum  = denom + (size_t)N_NODES * NUM_HEADS;

    const int nhd = N_NODES * HD;                        // 12.8M
    const int eh  = N_EDGES * NUM_HEADS;                 // 6.4M

    gat_init<<<(nhd + 255) / 256, 256, 0, stream>>>(mmax, denom, asum, rst);

    dim3 gg(N_NODES / 16, HD / 64);                      // (3125, 4)
    gat_gemm_wmma<<<gg, 128, 0, stream>>>(feat, fc_w, fs);

    gat_edge_max <<<(eh + 255) / 256, 256, 0, stream>>>(edge_emb, e_feat, dst, mmax);
    gat_edge_exp <<<(eh + 255) / 256, 256, 0, stream>>>(edge_emb, e_feat, dst, mmax, denom, attn);
    gat_edge_norm<<<(eh + 255) / 256, 256, 0, stream>>>(dst, denom, attn, asum);

    const int agg_threads = N_EDGES * (HD / 4);          // 51.2M
    gat_aggregate<<<(agg_threads + 255) / 256, 256, 0, stream>>>(src, dst, attn, fs, rst);

    gat_finalize<<<(nhd + 255) / 256, 256, 0, stream>>>(rst, asum);
}